// latent_manifold_loss_57870389346403
// MI455X (gfx1250) — compile-verified
//
#include <hip/hip_runtime.h>
#include <math.h>

// gfx1250 wave32 WMMA operand types
typedef __attribute__((ext_vector_type(2))) float v2f;
typedef __attribute__((ext_vector_type(8))) float v8f;

#define D_DIM 1024
#define NEIGH 64

// Phase 1: partial column reductions  ws[chunk][d] = sum_{rows in chunk} (A[r][d]-sel[d])^2
// Each wave (32 lanes) owns 16 columns. Per 4-row step we build the f32 16x4 A-operand
// (documented layout: lanes 0-15 -> M=lane, VGPR0/1 = K0/K1; lanes 16-31 -> K2/K3)
// holding squared differences, multiply by an all-ones 4x16 B (layout-irrelevant),
// and let v_wmma_f32_16x16x4_f32 accumulate the per-column sums in the C matrix.
__global__ __launch_bounds__(256)
void lml_phase1(const float* __restrict__ A, const int* __restrict__ gidp,
                float* __restrict__ ws, int rowsPerChunk) {
    const int lane = threadIdx.x & 31;
    const int wave = threadIdx.x >> 5;
    const int hi   = lane >> 4;          // 0: K=0,1   1: K=2,3
    const int c0   = blockIdx.x * 128 + wave * 16;   // this wave's 16-column group
    const int col  = c0 + (lane & 15);               // M = lane & 15

    const int gid = gidp[0];
    const float s = A[(size_t)gid * D_DIM + col];    // sel[col], per-lane constant

    // first row this lane touches inside the chunk
    const int r0 = blockIdx.y * rowsPerChunk + 2 * hi;
    const float* p = A + (size_t)r0 * D_DIM + col;

    v8f acc = {};                 // C/D accumulator, 16x16 f32 (8 VGPRs)
    v2f ones = {1.0f, 1.0f};      // B = all ones -> C[m][n] = sum_k A[m][k] for every n

    const int iters = rowsPerChunk >> 2;   // 4 rows consumed per WMMA
#pragma unroll 4
    for (int i = 0; i < iters; ++i) {
        float x0 = p[0];          // row r + 2*hi
        float x1 = p[D_DIM];      // row r + 2*hi + 1
        p += 4 * D_DIM;
        float t0 = x0 - s;
        float t1 = x1 - s;
        v2f q;
        q.x = t0 * t0;            // A[m][2*hi]
        q.y = t1 * t1;            // A[m][2*hi+1]
        acc = __builtin_amdgcn_wmma_f32_16x16x4_f32(
            /*neg_a=*/false, q, /*neg_b=*/false, ones,
            /*c_mod=*/(short)0, acc, /*reuse_a=*/false, /*reuse_b=*/false);
    }

    // All columns of C are identical; take column N=0.
    // Documented C layout: VGPR r, lane 0  -> (M=r,   N=0)  => cols c0+0..7
    //                      VGPR r, lane 16 -> (M=8+r, N=0)  => cols c0+8..15
    if ((lane & 15) == 0) {
        float* dst = ws + (size_t)blockIdx.y * D_DIM + c0 + hi * 8;
        *(v8f*)dst = acc;         // 32B-aligned (c0 multiple of 16 floats)
    }
}

// Phase 2: reduce chunk partials, sqrt, deterministic top-64-smallest mean.
__global__ __launch_bounds__(1024)
void lml_phase2(const float* __restrict__ ws, float* __restrict__ out, int R) {
    __shared__ float dist[D_DIM];
    __shared__ float sel[NEIGH];

    const int d = threadIdx.x;    // one thread per feature dimension
    float a = 0.0f;
    for (int c = 0; c < R; ++c)   // fixed order -> deterministic
        a += ws[(size_t)c * D_DIM + d];
    float dv = sqrtf(a);
    dist[d] = dv;
    __syncthreads();

    // rank by counting (ties broken by index -> ranks are a permutation of 0..1023)
    int cnt = 0;
    for (int j = 0; j < D_DIM; ++j) {
        float o = dist[j];
        cnt += (o < dv) || ((o == dv) && (j < d));
    }
    if (cnt < NEIGH) sel[cnt] = dv;   // unique slot per winner
    __syncthreads();

    if (d == 0) {
        float ssum = 0.0f;
        for (int i = 0; i < NEIGH; ++i) ssum += sel[i];   // fixed order
        out[0] = ssum * (1.0f / NEIGH);
    }
}

extern "C" void kernel_launch(void* const* d_in, const int* in_sizes, int n_in,
                              void* d_out, int out_size, void* d_ws, size_t ws_size,
                              hipStream_t stream) {
    const int*   gidp = (const int*)d_in[0];    // group_id (scalar int)
    const float* A    = (const float*)d_in[1];  // all_latents [N, 1024] f32
    float*       out  = (float*)d_out;          // scalar f32
    float*       ws   = (float*)d_ws;

    const long long total = in_sizes[1];
    const int N = (int)(total / D_DIM);         // 65536

    // Row-chunk count: enough waves to fill the GPU, bounded by scratch size.
    int R = 128;                                // 512 KB of partials
    while ((size_t)R * D_DIM * sizeof(float) > ws_size && R > 1) R >>= 1;
    if (R > N / 4) R = N / 4;                   // need >= 4 rows per chunk
    const int rowsPerChunk = N / R;

    dim3 grid(D_DIM / 128, R);                  // 8 column-blocks x R row-chunks
    lml_phase1<<<grid, 256, 0, stream>>>(A, gidp, ws, rowsPerChunk);
    lml_phase2<<<1, 1024, 0, stream>>>(ws, out, R);
}